// SAGPool_81243601371621
// MI455X (gfx1250) — compile-verified
//
#include <hip/hip_runtime.h>
#include <hip/hip_bf16.h>
#include <math.h>

// ---------------------------------------------------------------------------
// SAGPool GNN for MI455X (gfx1250, wave32).
// Dense GraphConv GEMMs -> V_WMMA_F32_16X16X4_F32 with LDS-staged, fragment-
// swizzled weights (ds_load_b64 per WMMA). Sparse edge aggregation ->
// global_atomic_add_f32 (memory-bound, ~4GB total, the roofline limiter).
// ---------------------------------------------------------------------------

typedef float v2f __attribute__((ext_vector_type(2)));
typedef float v8f __attribute__((ext_vector_type(8)));

#define NT      65536      // total nodes
#define NE      1048576    // total edges
#define IN_DIM  64
#define H_DIM   128
#define NGRAPH  128
#define NPG0    512        // nodes per graph before pooling
#define KEEP    410        // ceil(0.8 * 512)
#define NT2     (NGRAPH * KEEP)  // 52480 nodes after SAGPool

// -------------------------------- utility ----------------------------------

__global__ void fill_f32(float* __restrict__ p, float v, size_t n) {
  size_t i = (size_t)blockIdx.x * blockDim.x + threadIdx.x;
  size_t stride = (size_t)gridDim.x * blockDim.x;
  for (; i < n; i += stride) p[i] = v;
}

// ----------------------- sparse edge aggregation ---------------------------
// s[dst][d] += x[src][d] for every active edge (src>=0 && dst>=0).
// One thread per (edge, feature) element; D is a power of two (64 or 128).
__global__ void edge_gather_add(const int* __restrict__ src,
                                const int* __restrict__ dst,
                                const float* __restrict__ x,
                                float* __restrict__ s,
                                int logD, size_t total) {
  const int dmask = (1 << logD) - 1;
  size_t i = (size_t)blockIdx.x * blockDim.x + threadIdx.x;
  size_t stride = (size_t)gridDim.x * blockDim.x;
  for (; i < total; i += stride) {
    size_t e = i >> logD;
    int d = (int)(i & (size_t)dmask);
    int ss = src[e], dd = dst[e];
    if (ss >= 0 && dd >= 0)
      atomicAdd(s + (((size_t)dd) << logD) + d, x[(((size_t)ss) << logD) + d]);
  }
}

// cnt[dst] += 1 for active edges (in-degree for the mean).
__global__ void edge_count(const int* __restrict__ src,
                           const int* __restrict__ dst,
                           float* __restrict__ cnt, size_t ne) {
  size_t e = (size_t)blockIdx.x * blockDim.x + threadIdx.x;
  size_t stride = (size_t)gridDim.x * blockDim.x;
  for (; e < ne; e += stride) {
    int ss = src[e], dd = dst[e];
    if (ss >= 0 && dd >= 0) atomicAdd(cnt + dd, 1.0f);
  }
}

// score[dst] += t[src]   (add-aggregated GraphConv(H,1) via linearity)
__global__ void edge_score_add(const int* __restrict__ src,
                               const int* __restrict__ dst,
                               const float* __restrict__ t,
                               float* __restrict__ score, size_t ne) {
  size_t e = (size_t)blockIdx.x * blockDim.x + threadIdx.x;
  size_t stride = (size_t)gridDim.x * blockDim.x;
  for (; e < ne; e += stride) {
    int ss = src[e], dd = dst[e];
    if (ss >= 0 && dd >= 0) atomicAdd(score + dd, t[ss]);
  }
}

// t[i] = <h[i], pw_rel>,  score[i] = <h[i], pw_root> + pb   (b128 loads)
__global__ void node_score(const float* __restrict__ h,
                           const float* __restrict__ pwrel,
                           const float* __restrict__ pwroot,
                           const float* __restrict__ pb,
                           float* __restrict__ t,
                           float* __restrict__ score, int n) {
  int i = blockIdx.x * blockDim.x + threadIdx.x;
  if (i >= n) return;
  const float4* hp = (const float4*)(h + (size_t)i * H_DIM);
  const float4* wr = (const float4*)pwrel;
  const float4* wo = (const float4*)pwroot;
  float a = 0.f, b = 0.f;
  for (int k = 0; k < H_DIM / 4; ++k) {
    float4 v = hp[k];
    float4 r = wr[k];
    float4 o = wo[k];
    a += v.x * r.x + v.y * r.y + v.z * r.z + v.w * r.w;
    b += v.x * o.x + v.y * o.y + v.z * o.z + v.w * o.w;
  }
  t[i] = a;
  score[i] = b + pb[0];
}

// ------------------------------ SAGPool select -----------------------------
// One block (512 threads) per graph. Exact rank via LDS all-pairs compare
// (tie-break by index == jax.lax.top_k). Kept node with rank r gets compact
// id g*KEEP + r. Final output is invariant to kept-node ordering, so rank
// order is a valid (and reference-matching) permutation.
__global__ void sag_select(const float* __restrict__ score,
                           int* __restrict__ perm_src,
                           float* __restrict__ gain,
                           int* __restrict__ node_map) {
  __shared__ float sc[NPG0];
  int g = blockIdx.x;
  int tid = threadIdx.x;
  int gid = g * NPG0 + tid;
  float my = score[gid];
  sc[tid] = my;
  __syncthreads();
  int rank = 0;
  for (int j = 0; j < NPG0; ++j) {
    float sj = sc[j];
    rank += (sj > my) || (sj == my && j < tid);
  }
  if (rank < KEEP) {
    int nid = g * KEEP + rank;
    perm_src[nid] = gid;
    gain[nid] = tanhf(my);
    node_map[gid] = nid;
  } else {
    node_map[gid] = -1;
  }
}

// h_out[j][d] = h_in[perm_src[j]][d] * gain[j]
__global__ void pool_gather(const float* __restrict__ h,
                            const int* __restrict__ perm_src,
                            const float* __restrict__ gain,
                            float* __restrict__ out, size_t total) {
  size_t i = (size_t)blockIdx.x * blockDim.x + threadIdx.x;
  size_t stride = (size_t)gridDim.x * blockDim.x;
  for (; i < total; i += stride) {
    size_t node = i >> 7;            // H_DIM == 128
    int d = (int)(i & 127);
    out[i] = h[(((size_t)perm_src[node]) << 7) + d] * gain[node];
  }
}

// src2/dst2 = node_map[src/dst]; dropped nodes become -1 (edge masked).
__global__ void edge_remap(const int* __restrict__ src,
                           const int* __restrict__ dst,
                           const int* __restrict__ node_map,
                           int* __restrict__ src2, int* __restrict__ dst2,
                           size_t ne) {
  size_t e = (size_t)blockIdx.x * blockDim.x + threadIdx.x;
  size_t stride = (size_t)gridDim.x * blockDim.x;
  for (; e < ne; e += stride) {
    src2[e] = node_map[src[e]];
    dst2[e] = node_map[dst[e]];
  }
}

// global mean pool: one block (128 threads) per graph, thread d owns column d.
__global__ void graph_mean_pool(const float* __restrict__ h,
                                float* __restrict__ xcat,
                                int npg, int col_off) {
  int g = blockIdx.x;
  int d = threadIdx.x;  // 0..127
  const float* p = h + ((size_t)g * npg) * H_DIM + d;
  float s = 0.f;
  for (int r = 0; r < npg; ++r) s += p[(size_t)r * H_DIM];
  xcat[(size_t)g * (4 * H_DIM) + col_off + d] = s / (float)npg;
}

// ------------------------------- WMMA GEMM ---------------------------------
// out = act( (A1 .* 1/max(cnt,1)) @ W1 + A2 @ W2 + bias ), output width 128.
// V_WMMA_F32_16X16X4_F32: A is 16x4 (row = lane&15, K-half split at lane 16),
// B is 4x16 (col = lane&15, same K-half split), C/D 16x16 in 8 VGPRs.
//
// W is staged per-workgroup into a 64KB LDS chunk (K<=128 at a time) in a
// fragment-swizzled pair layout: pair slot holds (W[k][c], W[k+1][c]) so each
// WMMA's B fragment is one ds_load_b64. The kh1 (lanes 16-31) half is rotated
// by 16 pair slots so the two lane halves hit disjoint bank sets.
__device__ __forceinline__ v8f wmma_f32_16x16x4(v2f a, v2f b, v8f c) {
  return __builtin_amdgcn_wmma_f32_16x16x4_f32(
      /*neg_a=*/false, a, /*neg_b=*/false, b,
      /*c_mod=*/(short)0, c, /*reuse_a=*/false, /*reuse_b=*/false);
}

__global__ void __launch_bounds__(256)
gemm_dual_relu(const float* __restrict__ A1, const float* __restrict__ cnt,
               const float* __restrict__ A2,
               const float* __restrict__ W1, const float* __restrict__ W2,
               const float* __restrict__ bias, float* __restrict__ out,
               int n, int K, int do_relu) {
  __shared__ float lds[H_DIM * 128];   // 64 KB: one K<=128 chunk of W
  v2f* ldsp = (v2f*)lds;               // pair view

  const int tid  = threadIdx.x;
  const int wave = (blockIdx.x * blockDim.x + tid) >> 5;
  const int lane = tid & 31;
  const int mtiles = n >> 4;           // n is always a multiple of 16
  const bool active = wave < mtiles;   // uniform per wave

  const int col15 = lane & 15;
  const int row   = (wave << 4) + col15;   // A row owned by this lane
  const int khalf = (lane >> 4) << 1;      // 0 (lanes 0-15) or 2 (lanes 16-31)
  const int kh1   = khalf >> 1;            // 0 or 1

  float invc = 1.0f;
  if (active && cnt) invc = 1.0f / fmaxf(cnt[row], 1.0f);

  v8f acc[8] = {};                     // 8 N-tiles x 16x16 f32 accumulators

  for (int prod = 0; prod < 2; ++prod) {
    const float* A = (prod == 0) ? A1 : A2;
    const float* W = (prod == 0) ? W1 : W2;
    if (!A) break;
    const float scale = (prod == 0) ? invc : 1.0f;

    for (int kc = 0; kc < K; kc += 128) {
      const int KC = (K - kc < 128) ? (K - kc) : 128;

      // ---- cooperative fill of the swizzled W chunk ----
      for (int p = tid; p < KC * H_DIM; p += 256) {
        int k = p >> 7;          // 0..KC-1
        int c = p & 127;
        int kh = (k >> 1) & 1;   // which K-half of the quad
        int pr = ((k >> 2) << 8) + (kh << 7) + ((c + (kh << 4)) & 127);
        lds[(pr << 1) + (k & 1)] = W[(size_t)(kc + k) * H_DIM + c];
      }
      __syncthreads();

      if (active) {
        const float* ap = A + (size_t)row * K + kc + khalf;
        __builtin_prefetch(ap + 128, 0, 3);   // next chunk -> global_prefetch_b8
        for (int k0 = 0; k0 < KC; k0 += 4) {
          v2f a = *(const v2f*)(ap + k0);     // global_load_b64
          a.x *= scale;
          a.y *= scale;
          const v2f* bp = ldsp + ((k0 >> 2) << 8) + (kh1 << 7) + col15;
#pragma unroll
          for (int t = 0; t < 8; ++t) {
            int tt = (t + kh1) & 7;           // bank-rotation of the kh1 half
            v2f b = bp[tt << 4];              // ds_load_b64
            acc[t] = wmma_f32_16x16x4(a, b, acc[t]);
          }
        }
      }
      __syncthreads();
    }
  }

  if (!active) return;

  // C/D layout: VGPR r -> (M = base + r + (lane<16 ? 0 : 8), N = t*16 + col15)
  int mrow = (wave << 4) + ((lane >> 4) << 3);
#pragma unroll
  for (int t = 0; t < 8; ++t) {
    float bv = bias ? bias[t * 16 + col15] : 0.0f;
#pragma unroll
    for (int r = 0; r < 8; ++r) {
      float v = acc[t][r] + bv;
      if (do_relu) v = fmaxf(v, 0.0f);
      out[(size_t)(mrow + r) * H_DIM + t * 16 + col15] = v;
    }
  }
}

// ------------------------------- MLP head ----------------------------------
__global__ void mlp_head(const float* __restrict__ hidden,
                         const float* __restrict__ W2,
                         const float* __restrict__ b2,
                         float* __restrict__ out) {
  int g = blockIdx.x * blockDim.x + threadIdx.x;
  if (g >= NGRAPH) return;
  const float* hp = hidden + (size_t)g * H_DIM;
  float o0 = b2[0], o1 = b2[1];
  for (int k = 0; k < H_DIM; ++k) {
    float v = hp[k];
    o0 += v * W2[2 * k];
    o1 += v * W2[2 * k + 1];
  }
  float m = fmaxf(o0, o1);
  float lse = m + logf(expf(o0 - m) + expf(o1 - m));
  out[2 * g] = o0 - lse;
  out[2 * g + 1] = o1 - lse;
}

// ------------------------------ orchestration ------------------------------

extern "C" void kernel_launch(void* const* d_in, const int* in_sizes, int n_in,
                              void* d_out, int out_size, void* d_ws, size_t ws_size,
                              hipStream_t stream) {
  (void)in_sizes; (void)n_in; (void)out_size; (void)ws_size;

  const float* x        = (const float*)d_in[0];
  const int*   eidx     = (const int*)  d_in[1];
  // d_in[2] = batch (unused: graphs are equal-size, like the reference reshape)
  const float* W1_rel   = (const float*)d_in[3];
  const float* W1_root  = (const float*)d_in[4];
  const float* b1       = (const float*)d_in[5];
  const float* Wc_rel   = (const float*)d_in[6];
  const float* Wc_root  = (const float*)d_in[7];
  const float* bc       = (const float*)d_in[8];
  const float* pw_rel   = (const float*)d_in[9];
  const float* pw_root  = (const float*)d_in[10];
  const float* pb       = (const float*)d_in[11];
  const float* lin1_W   = (const float*)d_in[12];
  const float* lin1_b   = (const float*)d_in[13];
  const float* lin2_W   = (const float*)d_in[14];
  const float* lin2_b   = (const float*)d_in[15];
  float* outp = (float*)d_out;

  const int* src = eidx;           // edge_index[0]
  const int* dst = eidx + NE;      // edge_index[1]

  // ---- workspace layout (buffers reused across stages, ~106 MB) ----
  char* w = (char*)d_ws;
  float* s    = (float*)w; w += (size_t)NT * H_DIM * 4;   // aggregation sum
  float* buf0 = (float*)w; w += (size_t)NT * H_DIM * 4;   // h1 -> h3 -> h5
  float* buf1 = (float*)w; w += (size_t)NT * H_DIM * 4;   // h2 -> h4
  float* cnt  = (float*)w; w += (size_t)NT * 4;
  float* tbuf = (float*)w; w += (size_t)NT * 4;
  float* scr  = (float*)w; w += (size_t)NT * 4;
  int*   perm = (int*)  w; w += (size_t)NT2 * 4;
  float* gain = (float*)w; w += (size_t)NT2 * 4;
  int*   nmap = (int*)  w; w += (size_t)NT * 4;
  int*   src2 = (int*)  w; w += (size_t)NE * 4;
  int*   dst2 = (int*)  w; w += (size_t)NE * 4;
  float* xcat = (float*)w; w += (size_t)NGRAPH * 4 * H_DIM * 4;
  float* mlph = (float*)w; w += (size_t)NGRAPH * H_DIM * 4;

  const dim3 blk(256);
  const int FILL_G = 4096, EDGE_G = 4096, AGG_G = 8192;
  auto gemm_grid = [](int n) { return dim3((unsigned)(((n >> 4) + 7) / 8)); };

  // ---- conv1: GraphConv(IN=64 -> H, mean) ----
  fill_f32<<<FILL_G, blk, 0, stream>>>(s, 0.f, (size_t)NT * IN_DIM);
  fill_f32<<<FILL_G, blk, 0, stream>>>(cnt, 0.f, (size_t)NT);
  edge_gather_add<<<AGG_G, blk, 0, stream>>>(src, dst, x, s, 6, (size_t)NE * IN_DIM);
  edge_count<<<EDGE_G, blk, 0, stream>>>(src, dst, cnt, (size_t)NE);
  gemm_dual_relu<<<gemm_grid(NT), blk, 0, stream>>>(s, cnt, x, W1_rel, W1_root, b1,
                                                    buf0, NT, IN_DIM, 1);
  graph_mean_pool<<<NGRAPH, H_DIM, 0, stream>>>(buf0, xcat, NPG0, 0);

  // ---- conv2: GraphConv(H -> H, mean), Wc[0] ----
  fill_f32<<<FILL_G, blk, 0, stream>>>(s, 0.f, (size_t)NT * H_DIM);
  fill_f32<<<FILL_G, blk, 0, stream>>>(cnt, 0.f, (size_t)NT);
  edge_gather_add<<<AGG_G, blk, 0, stream>>>(src, dst, buf0, s, 7, (size_t)NE * H_DIM);
  edge_count<<<EDGE_G, blk, 0, stream>>>(src, dst, cnt, (size_t)NE);
  gemm_dual_relu<<<gemm_grid(NT), blk, 0, stream>>>(s, cnt, buf0, Wc_rel, Wc_root, bc,
                                                    buf1, NT, H_DIM, 1);
  graph_mean_pool<<<NGRAPH, H_DIM, 0, stream>>>(buf1, xcat, NPG0, H_DIM);

  // ---- SAGPooling: score = add-aggr GraphConv(H,1), keep top-410/graph ----
  node_score<<<(NT + 255) / 256, blk, 0, stream>>>(buf1, pw_rel, pw_root, pb,
                                                   tbuf, scr, NT);
  edge_score_add<<<EDGE_G, blk, 0, stream>>>(src, dst, tbuf, scr, (size_t)NE);
  sag_select<<<NGRAPH, NPG0, 0, stream>>>(scr, perm, gain, nmap);
  pool_gather<<<AGG_G, blk, 0, stream>>>(buf1, perm, gain, buf0,
                                         (size_t)NT2 * H_DIM);   // h3 -> buf0
  edge_remap<<<EDGE_G, blk, 0, stream>>>(src, dst, nmap, src2, dst2, (size_t)NE);

  // ---- conv3: Wc[1] on pooled graph (n = 52480, masked edges skipped) ----
  fill_f32<<<FILL_G, blk, 0, stream>>>(s, 0.f, (size_t)NT2 * H_DIM);
  fill_f32<<<FILL_G, blk, 0, stream>>>(cnt, 0.f, (size_t)NT2);
  edge_gather_add<<<AGG_G, blk, 0, stream>>>(src2, dst2, buf0, s, 7, (size_t)NE * H_DIM);
  edge_count<<<EDGE_G, blk, 0, stream>>>(src2, dst2, cnt, (size_t)NE);
  gemm_dual_relu<<<gemm_grid(NT2), blk, 0, stream>>>(
      s, cnt, buf0, Wc_rel + 1 * H_DIM * H_DIM, Wc_root + 1 * H_DIM * H_DIM,
      bc + 1 * H_DIM, buf1, NT2, H_DIM, 1);
  graph_mean_pool<<<NGRAPH, H_DIM, 0, stream>>>(buf1, xcat, KEEP, 2 * H_DIM);

  // ---- conv4: Wc[2] ----
  fill_f32<<<FILL_G, blk, 0, stream>>>(s, 0.f, (size_t)NT2 * H_DIM);
  fill_f32<<<FILL_G, blk, 0, stream>>>(cnt, 0.f, (size_t)NT2);
  edge_gather_add<<<AGG_G, blk, 0, stream>>>(src2, dst2, buf1, s, 7, (size_t)NE * H_DIM);
  edge_count<<<EDGE_G, blk, 0, stream>>>(src2, dst2, cnt, (size_t)NE);
  gemm_dual_relu<<<gemm_grid(NT2), blk, 0, stream>>>(
      s, cnt, buf1, Wc_rel + 2 * H_DIM * H_DIM, Wc_root + 2 * H_DIM * H_DIM,
      bc + 2 * H_DIM, buf0, NT2, H_DIM, 1);
  graph_mean_pool<<<NGRAPH, H_DIM, 0, stream>>>(buf0, xcat, KEEP, 3 * H_DIM);

  // ---- head: lin1 (WMMA GEMM, K=512, 4 LDS chunks) -> lin2 + log_softmax ----
  gemm_dual_relu<<<gemm_grid(NGRAPH), blk, 0, stream>>>(
      xcat, nullptr, nullptr, lin1_W, nullptr, lin1_b, mlph, NGRAPH, 4 * H_DIM, 1);
  mlp_head<<<1, H_DIM, 0, stream>>>(mlph, lin2_W, lin2_b, outp);
}